// Emu3VisionVQVectorQuantizer_10307921510587
// MI455X (gfx1250) — compile-verified
//
#include <hip/hip_runtime.h>
#include <hip/hip_bf16.h>

typedef __attribute__((ext_vector_type(2))) float v2f;
typedef __attribute__((ext_vector_type(8))) float v8f;
typedef int v4i_vs __attribute__((vector_size(16)));   // matches builtin param type

#define K_CODES  32768
#define N_VECS   16384
#define HW       4096      // H*W = 64*64
#define ZQ_ELEMS 65536     // B*T*C*H*W

#define KSPLIT   4         // K partitions across blockIdx.y
#define KRANGE   (K_CODES / KSPLIT)   // 8192 codes per partition
#define CHUNK    512       // E rows staged in LDS per buffer (8KB)
#define NCHUNKS  (KRANGE / CHUNK)     // 16

#if defined(__gfx1250__) && __has_builtin(__builtin_amdgcn_global_load_async_to_lds_b128)
#define USE_ASYNC_LDS 1
#endif

__device__ __forceinline__ void async_copy_row(const float4* gsrc, float4* ldst) {
#if defined(USE_ASYNC_LDS)
    // gfx1250 async DMA global->LDS (ASYNCcnt), b128 per row
    __builtin_amdgcn_global_load_async_to_lds_b128(
        (v4i_vs*)gsrc, (v4i_vs*)ldst, /*imm offset*/0, /*cpol*/0);
#else
    *ldst = *gsrc;   // fallback: VGPR bounce (global_load_b128 + ds_store_b128)
#endif
}

__device__ __forceinline__ void async_wait0() {
#if defined(USE_ASYNC_LDS)
#if __has_builtin(__builtin_amdgcn_s_wait_asynccnt)
    __builtin_amdgcn_s_wait_asynccnt(0);
#else
    asm volatile("s_wait_asynccnt 0" ::: "memory");
#endif
#endif
}

// Block = 8 waves = 8 x-tiles, all scanning the same K partition so E chunks
// can be shared through LDS. One wave handles a 16-row x-tile via
// V_WMMA_F32_16X16X4_F32:  D = (-2*X)(16x4) * E_chunk^T(4x16)  (C = inline 0),
// then d'[m][n] = D + |e_n|^2  (argmin-equivalent; |x|^2 row-constant dropped).
__global__ __launch_bounds__(256) void vq_partial_kernel(
    const float* __restrict__ x,      // (1,4,4,64,64) f32
    const float* __restrict__ E,      // (32768,4) f32
    float* __restrict__ pval,         // [n][p] partial min distance'
    int*   __restrict__ pidx)         // [n][p] partial argmin index
{
    __shared__ float4 ebuf[2][CHUNK];            // 16KB double buffer

    const int lane = threadIdx.x & 31;
    const int wave = threadIdx.x >> 5;
    const int tile = blockIdx.x * 8 + wave;      // 0..1023
    const int p    = blockIdx.y;                 // 0..3 K-partition
    const int kbase = p * KRANGE;
    const int n0   = tile << 4;
    const int half = lane >> 4;                  // 0: K=0,1 rows 0-7 ; 1: K=2,3 rows 8-15
    const int sub  = lane & 15;

    const float4* __restrict__ E4 = reinterpret_cast<const float4*>(E);

    // ---- A matrix (16x4 f32), pre-scaled by -2; ISA layout: v0=K(2h), v1=K(2h+1), M=sub ----
    v2f A;
    {
        const int n  = n0 + sub;
        const int t  = n >> 12;
        const int hw = n & (HW - 1);
        A.x = -2.0f * x[(t * 4 + 2 * half + 0) * HW + hw];
        A.y = -2.0f * x[(t * 4 + 2 * half + 1) * HW + hw];
    }

    float minv[8];
    int   mini[8];
#pragma unroll
    for (int r = 0; r < 8; ++r) { minv[r] = 3.402823e38f; mini[r] = 0; }

    // ---- stage chunk 0 (each of 256 threads DMAs 2 rows of 16B) ----
    {
        const int r0 = threadIdx.x * 2;
        async_copy_row(E4 + kbase + r0 + 0, &ebuf[0][r0 + 0]);
        async_copy_row(E4 + kbase + r0 + 1, &ebuf[0][r0 + 1]);
    }
    async_wait0();
    __syncthreads();

    for (int c = 0; c < NCHUNKS; ++c) {
        const int cur = c & 1;
        // prefetch next chunk into the other buffer (overlaps with compute below)
        if (c + 1 < NCHUNKS) {
            const int r0 = threadIdx.x * 2;
            const float4* src = E4 + kbase + (c + 1) * CHUNK;
            async_copy_row(src + r0 + 0, &ebuf[cur ^ 1][r0 + 0]);
            async_copy_row(src + r0 + 1, &ebuf[cur ^ 1][r0 + 1]);
        }

        const int kc = kbase + c * CHUNK;
#pragma unroll 4
        for (int j = 0; j < CHUNK; j += 16) {
            const float4 e = ebuf[cur][j + sub];           // ds_load_b128
            const float ee = e.x * e.x + e.y * e.y + e.z * e.z + e.w * e.w;

            // B layout: v0 = row K=(2*half), v1 = row K=(2*half+1), col N = sub
            v2f Bv;
            Bv.x = half ? e.z : e.x;
            Bv.y = half ? e.w : e.y;

            v8f Cz = {};   // folds to inline SRC2=0 -> no C-broadcast movs
            v8f D = __builtin_amdgcn_wmma_f32_16x16x4_f32(
                false, A, false, Bv, (short)0, Cz, false, false);

            const int idx = kc + j + sub;
#pragma unroll
            for (int r = 0; r < 8; ++r) {
                const float dv = D[r] + ee;              // off critical path of next WMMA
                const bool better = dv < minv[r];        // strict '<' keeps lowest k
                minv[r] = better ? dv  : minv[r];
                mini[r] = better ? idx : mini[r];
            }
        }

        async_wait0();      // staged chunk c+1 landed in LDS
        __syncthreads();    // everyone done reading ebuf[cur]
    }

    // ---- cross-lane argmin within each 16-lane half (masks < 16 stay in-half) ----
#pragma unroll
    for (int mask = 1; mask <= 8; mask <<= 1) {
#pragma unroll
        for (int r = 0; r < 8; ++r) {
            const float ov = __shfl_xor(minv[r], mask, 32);
            const int   oi = __shfl_xor(mini[r], mask, 32);
            const bool better = (ov < minv[r]) || ((ov == minv[r]) && (oi < mini[r]));
            minv[r] = better ? ov : minv[r];
            mini[r] = better ? oi : mini[r];
        }
    }

    // ---- lanes (half, sub=r<8) own row m = 8*half + r: write (val, idx) partial ----
    if (sub < 8) {
        float dmin = minv[0];
        int   code = mini[0];
#pragma unroll
        for (int r = 1; r < 8; ++r)
            if (sub == r) { dmin = minv[r]; code = mini[r]; }

        const int n = n0 + half * 8 + sub;
        pval[(n << 2) + p] = dmin;
        pidx[(n << 2) + p] = code;
    }
}

// Merge the KSPLIT partials per row, gather z_q, emit codes + block loss partials.
__global__ __launch_bounds__(256) void vq_finalize_kernel(
    const float* __restrict__ x,
    const float* __restrict__ E,
    const float* __restrict__ pval,
    const int*   __restrict__ pidx,
    float* __restrict__ out,
    float* __restrict__ lpart)
{
    const int n = blockIdx.x * 256 + threadIdx.x;    // 0..16383

    float bv = pval[n << 2];
    int   bi = pidx[n << 2];
#pragma unroll
    for (int p = 1; p < KSPLIT; ++p) {
        const float v = pval[(n << 2) + p];
        const int   i = pidx[(n << 2) + p];
        const bool better = (v < bv) || ((v == bv) && (i < bi));
        bv = better ? v : bv;
        bi = better ? i : bi;
    }

    const int t  = n >> 12;
    const int hw = n & (HW - 1);
    const float4 e = reinterpret_cast<const float4*>(E)[bi];
    const float z[4] = {e.x, e.y, e.z, e.w};

    float lsum = 0.0f;
#pragma unroll
    for (int ch = 0; ch < 4; ++ch) {
        const int xi = (t * 4 + ch) * HW + hw;
        const float xv = x[xi];
        out[xi] = z[ch];                       // z_q_st forward value == z_q
        const float dd = z[ch] - xv;
        lsum += dd * dd;
    }
    out[ZQ_ELEMS + 1 + n] = (float)bi;         // codes after the scalar loss slot

    __shared__ float sdata[256];
    sdata[threadIdx.x] = lsum;
    __syncthreads();
#pragma unroll
    for (int st = 128; st > 0; st >>= 1) {
        if ((int)threadIdx.x < st) sdata[threadIdx.x] += sdata[threadIdx.x + st];
        __syncthreads();
    }
    if (threadIdx.x == 0) lpart[blockIdx.x] = sdata[0];
}

__global__ __launch_bounds__(64) void vq_loss_kernel(
    const float* __restrict__ lpart, float* __restrict__ out)
{
    __shared__ float s[64];
    s[threadIdx.x] = lpart[threadIdx.x];
    __syncthreads();
#pragma unroll
    for (int st = 32; st > 0; st >>= 1) {
        if ((int)threadIdx.x < st) s[threadIdx.x] += s[threadIdx.x + st];
        __syncthreads();
    }
    if (threadIdx.x == 0)
        out[ZQ_ELEMS] = s[0] * (1.25f / 65536.0f);   // (1+BETA)*mean over 65536
}

extern "C" void kernel_launch(void* const* d_in, const int* in_sizes, int n_in,
                              void* d_out, int out_size, void* d_ws, size_t ws_size,
                              hipStream_t stream) {
    const float* x = (const float*)d_in[0];     // 65536 f32
    const float* E = (const float*)d_in[1];     // 131072 f32
    float* out     = (float*)d_out;             // 81921 f32

    char* ws = (char*)d_ws;
    float* pval  = (float*)(ws);                        // 65536 f32 (256KB)
    int*   pidx  = (int*)  (ws + (size_t)N_VECS * KSPLIT * 4);        // 65536 i32
    float* lpart = (float*)(ws + (size_t)N_VECS * KSPLIT * 8);        // 64 f32

    vq_partial_kernel<<<dim3(128, KSPLIT), 256, 0, stream>>>(x, E, pval, pidx);
    vq_finalize_kernel<<<64, 256, 0, stream>>>(x, E, pval, pidx, out, lpart);
    vq_loss_kernel<<<1, 64, 0, stream>>>(lpart, out);
}